// FlaxAttentionModule_76879914598682
// MI455X (gfx1250) — compile-verified
//
#include <hip/hip_runtime.h>
#include <hip/hip_bf16.h>
#include <stdint.h>

// Problem constants (from the reference)
#define B_      16
#define QLEN_   1
#define HKV_    8
#define D_      128
#define MAXLEN_ 8192
#define ROW_    (HKV_ * D_)                    // 1024 floats per (b, pos)
#define CACHE_N_ ((size_t)B_ * MAXLEN_ * ROW_) // 134,217,728 floats per cache

// Bulk-copy kernel: stream 16B chunks GMEM -> LDS -> GMEM via the CDNA5
// async DMA path. Each thread moves CPT float4's (64 B); a 256-thread block
// stages 16 KB in LDS. NT temporal hints keep the 512MB streams out of L2.
#define TPB 256
#define CPT 4

__global__ __launch_bounds__(TPB) void async_copy_kernel(const float4* __restrict__ src,
                                                         float4* __restrict__ dst) {
    __shared__ float4 stage[TPB * CPT];
    const unsigned tid = threadIdx.x;
    const unsigned long long base =
        (unsigned long long)blockIdx.x * (unsigned long long)(TPB * CPT);

    // Issue CPT async loads (ASYNCcnt += CPT), fully coalesced per wave.
#pragma unroll
    for (int i = 0; i < CPT; ++i) {
        unsigned long long idx = base + (unsigned long long)(i * TPB) + tid;
        unsigned lds_off = (unsigned)(uintptr_t)&stage[i * TPB + tid];
        unsigned long long gaddr = (unsigned long long)(uintptr_t)(src + idx);
        asm volatile("global_load_async_to_lds_b128 %0, %1, off th:TH_LOAD_NT"
                     :: "v"(lds_off), "v"(gaddr) : "memory");
    }
    // Drain this wave's async loads before reading LDS back out.
    asm volatile("s_wait_asynccnt 0" ::: "memory");

    // Issue CPT async stores LDS -> GMEM (non-temporal writes).
#pragma unroll
    for (int i = 0; i < CPT; ++i) {
        unsigned long long idx = base + (unsigned long long)(i * TPB) + tid;
        unsigned lds_off = (unsigned)(uintptr_t)&stage[i * TPB + tid];
        unsigned long long gaddr = (unsigned long long)(uintptr_t)(dst + idx);
        asm volatile("global_store_async_from_lds_b128 %0, %1, off th:TH_STORE_NT"
                     :: "v"(gaddr), "v"(lds_off) : "memory");
    }
    // Make sure stores are issued/complete before LDS is released (S_ENDPGM
    // also does an implicit wait-idle, but be explicit).
    asm volatile("s_wait_asynccnt 0" ::: "memory");
}

// Overwrite the dynamically-updated row r = end_index % MAXLEN in both caches.
// 16 blocks (one per batch), 2*1024 floats each — negligible traffic.
__global__ void update_rows_kernel(const float* __restrict__ newk,
                                   const float* __restrict__ newv,
                                   const int* __restrict__ end_index_p,
                                   float* __restrict__ out_kc,
                                   float* __restrict__ out_vc) {
    const int b = blockIdx.x;
    const int r = end_index_p[0] % MAXLEN_;
    const size_t dst = ((size_t)b * MAXLEN_ + (size_t)r) * ROW_;
    const size_t srcb = (size_t)b * ROW_; // new key/value: (B, 1, HKV, D)
    for (int i = threadIdx.x; i < ROW_; i += blockDim.x) {
        out_kc[dst + i] = newk[srcb + i];
        out_vc[dst + i] = newv[srcb + i];
    }
}

// mask[b,t] = attn[b,t] && causal[end_index, t] && (t < end_index + QLEN)
// bias      = mask ? 0 : float32 finfo.min (= -FLT_MAX)
__global__ void mask_kernel(const unsigned char* __restrict__ attn_mask,
                            const unsigned char* __restrict__ causal_mask,
                            const int* __restrict__ end_index_p,
                            float* __restrict__ out_mask,
                            float* __restrict__ out_bias) {
    const long long total = (long long)B_ * MAXLEN_;
    long long i = (long long)blockIdx.x * blockDim.x + threadIdx.x;
    if (i >= total) return;
    const int t = (int)(i % MAXLEN_);
    const int e = end_index_p[0];
    const bool am = attn_mask[i] != 0;
    const bool cm = causal_mask[(long long)e * MAXLEN_ + t] != 0;
    const bool ok = am && cm && (t < e + QLEN_);
    out_mask[i] = ok ? 1.0f : 0.0f;
    out_bias[i] = ok ? 0.0f : -3.40282346638528859812e+38f;
}

extern "C" void kernel_launch(void* const* d_in, const int* in_sizes, int n_in,
                              void* d_out, int out_size, void* d_ws, size_t ws_size,
                              hipStream_t stream) {
    (void)in_sizes; (void)n_in; (void)out_size; (void)d_ws; (void)ws_size;

    // setup_inputs() order:
    // 0 query (unused), 1 key, 2 value, 3 key_cache, 4 value_cache,
    // 5 attention_mask (bool), 6 causal_mask (bool), 7 end_index (int scalar)
    const float* newk        = (const float*)d_in[1];
    const float* newv        = (const float*)d_in[2];
    const float* key_cache   = (const float*)d_in[3];
    const float* value_cache = (const float*)d_in[4];
    const unsigned char* am  = (const unsigned char*)d_in[5];
    const unsigned char* cm  = (const unsigned char*)d_in[6];
    const int* end_index     = (const int*)d_in[7];

    // Outputs, concatenated flat in return order:
    float* out      = (float*)d_out;
    float* out_kc   = out;
    float* out_vc   = out_kc + CACHE_N_;
    float* out_mask = out_vc + CACHE_N_;
    float* out_bias = out_mask + (size_t)B_ * MAXLEN_;

    // Bulk cache copies: CACHE_N_/4 float4 chunks, TPB*CPT per block.
    const unsigned long long chunks = CACHE_N_ / 4;              // 33,554,432
    const unsigned blocks = (unsigned)(chunks / (TPB * CPT));    // 32,768 (exact)
    async_copy_kernel<<<blocks, TPB, 0, stream>>>((const float4*)key_cache,
                                                  (float4*)out_kc);
    async_copy_kernel<<<blocks, TPB, 0, stream>>>((const float4*)value_cache,
                                                  (float4*)out_vc);

    // Row update must land after the bulk copy — same stream guarantees order.
    update_rows_kernel<<<B_, 256, 0, stream>>>(newk, newv, end_index,
                                               out_kc, out_vc);

    const int mtot = B_ * MAXLEN_;
    mask_kernel<<<(mtot + 255) / 256, 256, 0, stream>>>(am, cm, end_index,
                                                        out_mask, out_bias);
}